// ChebGCN_19722489823541
// MI455X (gfx1250) — compile-verified
//
#include <hip/hip_runtime.h>
#include <hip/hip_bf16.h>

typedef __attribute__((ext_vector_type(2))) float v2f;
typedef __attribute__((ext_vector_type(8))) float v8f;

#define NNODES 10000

// ---------------- utility: zero a float4-aligned buffer ----------------
__global__ __launch_bounds__(256) void zero4_kernel(float4* __restrict__ p, int n4) {
    int i = blockIdx.x * 256 + threadIdx.x;
    int stride = gridDim.x * 256;
    float4 z; z.x = 0.f; z.y = 0.f; z.z = 0.f; z.w = 0.f;
    for (; i < n4; i += stride) p[i] = z;
}

// ---------------- transpose W[K][NOut] -> WT[NOut][K] ----------------
__global__ __launch_bounds__(256) void transpose_kernel(const float* __restrict__ in,
                                                        float* __restrict__ out,
                                                        int K, int NOut) {
    int idx = blockIdx.x * 256 + threadIdx.x;
    if (idx < K * NOut) {
        int k = idx / NOut;
        int c = idx - k * NOut;
        out[(size_t)c * K + k] = in[idx];
    }
}

// ---------------- degree via f32 atomics into dis buffer ----------------
__global__ __launch_bounds__(256) void deg_kernel(const int* __restrict__ src,
                                                  float* __restrict__ deg, int E) {
    int i = blockIdx.x * 256 + threadIdx.x;
    if (i < E) atomicAdd(&deg[src[i]], 1.0f);
}

// dis[i] = deg>0 ? rsqrt(deg) : 0   (deg is an integer count, so max(deg,1)=deg)
__global__ __launch_bounds__(256) void dis_kernel(float* __restrict__ deg, int n) {
    int i = blockIdx.x * 256 + threadIdx.x;
    if (i < n) {
        float d = deg[i];
        deg[i] = (d > 0.0f) ? rsqrtf(d) : 0.0f;
    }
}

// ---------------- edge scatter: agg[dst] += norm * x[src] ----------------
// One thread handles a 4-float chunk of one edge. shift = log2(D/4).
__global__ __launch_bounds__(256) void scatter_kernel(
    const float* __restrict__ xin, const int* __restrict__ src,
    const int* __restrict__ dst, const float* __restrict__ dis,
    float* __restrict__ agg, int E, int shift)
{
    unsigned tid = blockIdx.x * 256u + threadIdx.x;
    int e = (int)(tid >> shift);
    if (e >= E) return;
    int c = (int)(tid & ((1u << shift) - 1u));
    int s = src[e];
    int d = dst[e];
    float norm = -(dis[s] * dis[d]);
    int D = 4 << shift;
    const float4 v = *(const float4*)(xin + (size_t)s * D + 4 * c);
    float* outp = agg + (size_t)d * D + 4 * c;
    atomicAdd(outp + 0, norm * v.x);
    atomicAdd(outp + 1, norm * v.y);
    atomicAdd(outp + 2, norm * v.z);
    atomicAdd(outp + 3, norm * v.w);
}

// ---------------- fused Cheb GEMM: Out = Xin@W0 + Agg@W1 + b (+ReLU) ------
// V_WMMA_F32_16X16X4_F32; one wave32 computes a 16 x (16*TILES) strip so each
// A float2 (b64 load) feeds TILES WMMAs. Weights are pre-transposed to
// [NOut][K] so B operands are contiguous b64 loads too.
// Two-stage software pipeline: loads for k+4 are issued before the WMMAs of
// step k, so the per-step wait is partial and WMMAs overlap load latency.
//
// A (16x4, f32) ISA layout: lanes 0-15 row M=lane, VGPR0=K0 VGPR1=K1;
//                           lanes 16-31 row M=lane-16, VGPR0=K2 VGPR1=K3.
// B (4x16, f32) mirrored with N in place of M.
// C/D (16x16, f32, 8 VGPRs): VGPR r -> M=r (lanes 0-15, N=lane) and
//                                        M=r+8 (lanes 16-31, N=lane-16).
template <int TILES>
__global__ __launch_bounds__(128) void cheb_gemm_kernel(
    const float* __restrict__ Xin, const float* __restrict__ Agg,
    const float* __restrict__ W0T, const float* __restrict__ W1T,
    const float* __restrict__ bias, float* __restrict__ Out,
    int K, int NOut, int relu)
{
    const int wave  = threadIdx.x >> 5;
    const int lane  = threadIdx.x & 31;
    const int rlane = lane & 15;            // row (A) / col (B) within tile
    const int khalf = (lane >> 4) << 1;     // 0 or 2
    const int rowBase = blockIdx.x << 4;
    const int colBase = wave * (TILES * 16);

    const float* a0p = Xin + (size_t)(rowBase + rlane) * K + khalf;
    const float* a1p = Agg + (size_t)(rowBase + rlane) * K + khalf;
    const float* w0p[TILES];
    const float* w1p[TILES];
#pragma unroll
    for (int t = 0; t < TILES; ++t) {
        w0p[t] = W0T + (size_t)(colBase + t * 16 + rlane) * K + khalf;
        w1p[t] = W1T + (size_t)(colBase + t * 16 + rlane) * K + khalf;
    }

    v8f acc[TILES];
#pragma unroll
    for (int t = 0; t < TILES; ++t) acc[t] = (v8f){};

    // ---- pipeline stage registers ----
    v2f a0c, a1c, b0c[TILES], b1c[TILES];

    // prologue: load stage 0
    a0c = *(const v2f*)(a0p);
    a1c = *(const v2f*)(a1p);
#pragma unroll
    for (int t = 0; t < TILES; ++t) {
        b0c[t] = *(const v2f*)(w0p[t]);
        b1c[t] = *(const v2f*)(w1p[t]);
    }

#pragma unroll 2
    for (int k = 0; k < K - 4; k += 4) {
        // prefetch stage k+4 while computing stage k
        v2f a0n = *(const v2f*)(a0p + k + 4);
        v2f a1n = *(const v2f*)(a1p + k + 4);
        v2f b0n[TILES], b1n[TILES];
#pragma unroll
        for (int t = 0; t < TILES; ++t) b0n[t] = *(const v2f*)(w0p[t] + k + 4);
#pragma unroll
        for (int t = 0; t < TILES; ++t) b1n[t] = *(const v2f*)(w1p[t] + k + 4);

#pragma unroll
        for (int t = 0; t < TILES; ++t)
            acc[t] = __builtin_amdgcn_wmma_f32_16x16x4_f32(
                false, a0c, false, b0c[t], (short)0, acc[t], false, false);
#pragma unroll
        for (int t = 0; t < TILES; ++t)
            acc[t] = __builtin_amdgcn_wmma_f32_16x16x4_f32(
                false, a1c, false, b1c[t], (short)0, acc[t], false, false);

        a0c = a0n; a1c = a1n;
#pragma unroll
        for (int t = 0; t < TILES; ++t) { b0c[t] = b0n[t]; b1c[t] = b1n[t]; }
    }

    // epilogue: last stage
#pragma unroll
    for (int t = 0; t < TILES; ++t)
        acc[t] = __builtin_amdgcn_wmma_f32_16x16x4_f32(
            false, a0c, false, b0c[t], (short)0, acc[t], false, false);
#pragma unroll
    for (int t = 0; t < TILES; ++t)
        acc[t] = __builtin_amdgcn_wmma_f32_16x16x4_f32(
            false, a1c, false, b1c[t], (short)0, acc[t], false, false);

    const int rowOff = rowBase + ((lane >> 4) << 3);
#pragma unroll
    for (int t = 0; t < TILES; ++t) {
        const int col = colBase + t * 16 + rlane;
        const float bv = bias[col];
#pragma unroll
        for (int r = 0; r < 8; ++r) {
            float v = acc[t][r] + bv;
            if (relu) v = fmaxf(v, 0.0f);
            Out[(size_t)(rowOff + r) * NOut + col] = v;
        }
    }
}

extern "C" void kernel_launch(void* const* d_in, const int* in_sizes, int n_in,
                              void* d_out, int out_size, void* d_ws, size_t ws_size,
                              hipStream_t stream) {
    const float* x   = (const float*)d_in[0];
    const int*   ei  = (const int*)d_in[1];
    const float* W01 = (const float*)d_in[3];
    const float* W11 = (const float*)d_in[4];
    const float* b1  = (const float*)d_in[5];
    const float* W02 = (const float*)d_in[6];
    const float* W12 = (const float*)d_in[7];
    const float* b2  = (const float*)d_in[8];
    const float* W03 = (const float*)d_in[9];
    const float* W13 = (const float*)d_in[10];
    const float* b3  = (const float*)d_in[11];

    const int N = NNODES;
    const int E = in_sizes[1] / 2;
    const int* src = ei;
    const int* dst = ei + E;

    // workspace layout (floats)
    float* dis  = (float*)d_ws;                   // N (padded to 10240)
    float* agg  = dis + 10240;                    // N*256
    float* h1   = agg + (size_t)N * 256;          // N*256
    float* h2   = h1  + (size_t)N * 256;          // N*256
    float* w01t = h2  + (size_t)N * 256;          // 128*256
    float* w11t = w01t + 128 * 256;               // 128*256
    float* w02t = w11t + 128 * 256;               // 256*256
    float* w12t = w02t + 256 * 256;               // 256*256
    float* w03t = w12t + 256 * 256;               // 256*128
    float* w13t = w03t + 256 * 128;               // 256*128
    float* out  = (float*)d_out;

    // ---- weight transposes (independent; tiny) ----
    transpose_kernel<<<(128 * 256 + 255) / 256, 256, 0, stream>>>(W01, w01t, 128, 256);
    transpose_kernel<<<(128 * 256 + 255) / 256, 256, 0, stream>>>(W11, w11t, 128, 256);
    transpose_kernel<<<(256 * 256 + 255) / 256, 256, 0, stream>>>(W02, w02t, 256, 256);
    transpose_kernel<<<(256 * 256 + 255) / 256, 256, 0, stream>>>(W12, w12t, 256, 256);
    transpose_kernel<<<(256 * 128 + 255) / 256, 256, 0, stream>>>(W03, w03t, 256, 128);
    transpose_kernel<<<(256 * 128 + 255) / 256, 256, 0, stream>>>(W13, w13t, 256, 128);

    // ---- degree -> dis ----
    zero4_kernel<<<64, 256, 0, stream>>>((float4*)dis, N / 4);
    deg_kernel<<<(E + 255) / 256, 256, 0, stream>>>(src, dis, E);
    dis_kernel<<<(N + 255) / 256, 256, 0, stream>>>(dis, N);

    // ---- layer 1: x[N,128] -> h1[N,256], ReLU ----
    zero4_kernel<<<1024, 256, 0, stream>>>((float4*)agg, N * 128 / 4);
    {
        unsigned nth = (unsigned)E << 5;          // E * (128/4)
        scatter_kernel<<<(nth + 255) / 256, 256, 0, stream>>>(x, src, dst, dis, agg, E, 5);
    }
    cheb_gemm_kernel<4><<<N / 16, 128, 0, stream>>>(x, agg, w01t, w11t, b1, h1, 128, 256, 1);

    // ---- layer 2: h1[N,256] -> h2[N,256], ReLU ----
    zero4_kernel<<<2048, 256, 0, stream>>>((float4*)agg, N * 256 / 4);
    {
        unsigned nth = (unsigned)E << 6;          // E * (256/4)
        scatter_kernel<<<(nth + 255) / 256, 256, 0, stream>>>(h1, src, dst, dis, agg, E, 6);
    }
    cheb_gemm_kernel<4><<<N / 16, 128, 0, stream>>>(h1, agg, w02t, w12t, b2, h2, 256, 256, 1);

    // ---- layer 3: h2[N,256] -> out[N,128], no ReLU ----
    zero4_kernel<<<2048, 256, 0, stream>>>((float4*)agg, N * 256 / 4);
    {
        unsigned nth = (unsigned)E << 6;
        scatter_kernel<<<(nth + 255) / 256, 256, 0, stream>>>(h2, src, dst, dis, agg, E, 6);
    }
    cheb_gemm_kernel<2><<<N / 16, 128, 0, stream>>>(h2, agg, w03t, w13t, b3, out, 256, 128, 0);
}